// RelaxationLayerImplicit_38259568672785
// MI455X (gfx1250) — compile-verified
//
#include <hip/hip_runtime.h>
#include <math.h>

// RelaxationLayerImplicit on MI455X (gfx1250, wave32).
//
// Block-structured solve:
//   A = [[I, B], [-B^T, 2I]],  z1 = v0 + 0.1*q,  z2 = v1 + tanh(v0)
//   (2I + B^T B) y = z2 + B^T z1 ;  x = z1 - B y
// M = 2I + B^T B has lambda in [2, ~12.3] => Newton-Schulz inversion
// (X <- X(2I - M X), X0 = 0.137*I) converges in ~7 iterations; every step
// is an fp32 WMMA GEMM (V_WMMA_F32_16X16X4_F32), keeping full precision.

#define UNITS 1024
#define BATCH 8192

typedef float v2f __attribute__((ext_vector_type(2)));
typedef float v8f __attribute__((ext_vector_type(8)));

// ---------------------------------------------------------------- transpose
__global__ __launch_bounds__(256)
void transpose_kernel(const float* __restrict__ in, float* __restrict__ out, int n) {
  __shared__ float tile[32][33];
  int bx = blockIdx.x * 32, by = blockIdx.y * 32;
  int tx = threadIdx.x, ty = threadIdx.y;           // block (32,8)
  #pragma unroll
  for (int i = 0; i < 32; i += 8)
    tile[ty + i][tx] = in[(size_t)(by + ty + i) * n + bx + tx];
  __syncthreads();
  #pragma unroll
  for (int i = 0; i < 32; i += 8)
    out[(size_t)(bx + ty + i) * n + by + tx] = tile[tx][ty + i];
}

// ------------------------------------------------------------------ rhs
// z1 = v0 + 0.1*q[row];  r = v1 + tanh(v0)   (dt/eps == 1.0)
__global__ __launch_bounds__(256)
void rhs_kernel(const float* __restrict__ v0, const float* __restrict__ v1,
                const float* __restrict__ q, float* __restrict__ z1,
                float* __restrict__ r) {
  size_t idx = (size_t)blockIdx.x * blockDim.x + threadIdx.x;
  int u = (int)(idx >> 13);                         // / 8192
  float a = v0[idx];
  z1[idx] = a + 0.1f * q[u];
  r[idx]  = v1[idx] + tanhf(a);
}

// ---------------------------------------------------------------- X0 = a*I
__global__ __launch_bounds__(256)
void init_inv_kernel(float* __restrict__ X) {
  size_t idx = (size_t)blockIdx.x * blockDim.x + threadIdx.x;
  int row = (int)(idx >> 10), col = (int)(idx & 1023);
  X[idx] = (row == col) ? 0.1369863f : 0.0f;        // 2/(lmin+lmax) ~ 2/14.6
}

// ------------------------------------------------- generic fp32 WMMA GEMM
// C = alpha*A[MxK]*B[KxN] (+ beta*C0) (+ dg*I)   row-major, dims % BM/BN == 0.
// Wave tile 32x64 (2x4 WMMA tiles). Double-buffered LDS, b128 staging.
// B is staged K-pair-interleaved so a B-fragment {K,K+1} is ONE aligned
// ds_load_b64. Staging registers live only inside stage() (no residency
// across the WMMA block -> no scratch spills); global-load latency is
// covered by distance-2 global_prefetch_b8 into GL2.
template <int BM, int BN, bool BETA, bool DIAG>
__global__ __launch_bounds__((BM / 32) * (BN / 64) * 32, 1)
void gemm_wmma(const float* __restrict__ A, const float* __restrict__ B,
               const float* __restrict__ C0, float* __restrict__ C,
               int M, int N, int K, float alpha, float beta, float dg) {
  constexpr int BK      = 16;
  constexpr int WAVES_M = BM / 32;
  constexpr int WAVES_N = BN / 64;
  constexpr int NT      = WAVES_M * WAVES_N * 32;
  constexpr int LDA_S   = BK + 4;      // 20 words: 32 distinct banks, A frags
  constexpr int LDP     = 2 * BN + 32; // pair-row stride; LDP % 64 == 32
  constexpr int AITER   = (BM * BK / 4) / NT;
  constexpr int BITER   = (2 * BN) / NT;   // 2x4 micro-blocks per thread
  constexpr int BN4     = BN / 4;

  __shared__ float As[2][BM * LDA_S];
  __shared__ float Bs[2][(BK / 2) * LDP];

  const int tid  = threadIdx.x;
  const int lane = tid & 31;
  const int wave = tid >> 5;
  const int wm   = wave / WAVES_N;
  const int wn   = wave % WAVES_N;
  const int half = lane >> 4;        // lanes 16-31 hold K+2/K+3
  const int l16  = lane & 15;

  const int rowBase = blockIdx.y * BM;
  const int colBase = blockIdx.x * BN;

  const float* Aptr = A + (size_t)rowBase * K;
  const float* Bptr = B + colBase;

  v8f acc[2][4] = {};

  // load + LDS-store one k-chunk; registers die at return (no spills)
  auto stage = [&](int k0, int buf) {
    float4 aR[AITER], b0R[BITER], b1R[BITER];
    #pragma unroll
    for (int i = 0; i < AITER; i++) {
      int f = tid + i * NT; int r = f >> 2, c4 = f & 3;
      aR[i] = *(const float4*)(Aptr + (size_t)r * K + k0 + c4 * 4);
    }
    #pragma unroll
    for (int i = 0; i < BITER; i++) {
      int blk = tid + i * NT; int kp = blk / BN4, c4 = blk % BN4;
      const float* p = Bptr + (size_t)(k0 + 2 * kp) * N + c4 * 4;
      b0R[i] = *(const float4*)p;
      b1R[i] = *(const float4*)(p + N);
    }
    #pragma unroll
    for (int i = 0; i < AITER; i++) {
      int f = tid + i * NT; int r = f >> 2, c4 = f & 3;
      *(float4*)&As[buf][r * LDA_S + c4 * 4] = aR[i];
    }
    #pragma unroll
    for (int i = 0; i < BITER; i++) {
      int blk = tid + i * NT; int kp = blk / BN4, c4 = blk % BN4;
      float4 q0 = {b0R[i].x, b1R[i].x, b0R[i].y, b1R[i].y};
      float4 q1 = {b0R[i].z, b1R[i].z, b0R[i].w, b1R[i].w};
      int base = kp * LDP + c4 * 8;
      *(float4*)&Bs[buf][base]     = q0;
      *(float4*)&Bs[buf][base + 4] = q1;
    }
  };
  auto prefetch_chunk = [&](int k0) {
    #pragma unroll
    for (int i = 0; i < AITER; i++) {
      int f = tid + i * NT; int r = f >> 2, c4 = f & 3;
      __builtin_prefetch(Aptr + (size_t)r * K + k0 + c4 * 4, 0, 3);
    }
    #pragma unroll
    for (int i = 0; i < BITER; i++) {
      int blk = tid + i * NT; int kp = blk / BN4, c4 = blk % BN4;
      __builtin_prefetch(Bptr + (size_t)(k0 + 2 * kp) * N + c4 * 4, 0, 3);
    }
  };

  stage(0, 0);
  __syncthreads();

  const int nk = K / BK;
  for (int kc = 0; kc < nk; kc++) {
    const int cur = kc & 1;
    if (kc + 2 < nk) prefetch_chunk((kc + 2) * BK);
    if (kc + 1 < nk) stage((kc + 1) * BK, cur ^ 1);  // write other buffer:
                                                     // its readers passed the
                                                     // previous barrier
    #pragma unroll
    for (int ks = 0; ks < BK; ks += 4) {
      v2f af[2]; v2f bf[4];
      #pragma unroll
      for (int mt = 0; mt < 2; mt++) {
        int row = wm * 32 + mt * 16 + l16;
        af[mt] = *(const v2f*)&As[cur][row * LDA_S + ks + 2 * half];
      }
      #pragma unroll
      for (int nt = 0; nt < 4; nt++) {
        int col = wn * 64 + nt * 16 + l16;
        bf[nt] = *(const v2f*)&Bs[cur][((ks >> 1) + half) * LDP + col * 2];
      }
      #pragma unroll
      for (int mt = 0; mt < 2; mt++)
        #pragma unroll
        for (int nt = 0; nt < 4; nt++)
          acc[mt][nt] = __builtin_amdgcn_wmma_f32_16x16x4_f32(
              false, af[mt], false, bf[nt], (short)0, acc[mt][nt], false, false);
    }

    if (kc + 1 < nk) __syncthreads();
  }

  // epilogue: C/D layout -> VGPR v holds rows v (lanes 0-15) / v+8 (16-31)
  #pragma unroll
  for (int mt = 0; mt < 2; mt++)
    #pragma unroll
    for (int nt = 0; nt < 4; nt++) {
      int gr0 = rowBase + wm * 32 + mt * 16 + half * 8;
      int gc  = colBase + wn * 64 + nt * 16 + l16;
      size_t base = (size_t)gr0 * N + gc;
      #pragma unroll
      for (int v = 0; v < 8; v++) {
        float val = alpha * acc[mt][nt][v];
        if constexpr (BETA) val += beta * C0[base + (size_t)v * N];
        if constexpr (DIAG) { if (gr0 + v == gc) val += dg; }
        C[base + (size_t)v * N] = val;
      }
    }
}

// ---------------------------------------------------------------- driver
extern "C" void kernel_launch(void* const* d_in, const int* in_sizes, int n_in,
                              void* d_out, int out_size, void* d_ws, size_t ws_size,
                              hipStream_t stream) {
  const float* v0 = (const float*)d_in[0];   // (1024, 8192)
  const float* v1 = (const float*)d_in[1];   // (1024, 8192)
  const float* B0 = (const float*)d_in[2];   // (1024, 1024)
  const float* q  = (const float*)d_in[3];   // (1024,)

  float* outp = (float*)d_out;
  float* x = outp;                                // z[:u]
  float* y = outp + (size_t)UNITS * BATCH;        // z[u:]

  const size_t UU = (size_t)UNITS * UNITS;
  const size_t UB = (size_t)UNITS * BATCH;
  float* ws = (float*)d_ws;
  float* Bt = ws;            // B0^T          4 MB
  float* Mm = Bt + UU;       // 2I + B^T B    4 MB
  float* X  = Mm + UU;       // NS iterate    4 MB
  float* X2 = X  + UU;       // NS ping-pong  4 MB
  float* T  = X2 + UU;       // M*X           4 MB
  float* z1 = T  + UU;       // rhs top      32 MB
  float* r  = z1 + UB;       // rhs / Schur  32 MB

  transpose_kernel<<<dim3(UNITS / 32, UNITS / 32), dim3(32, 8), 0, stream>>>(B0, Bt, UNITS);
  rhs_kernel<<<(unsigned)(UB / 256), 256, 0, stream>>>(v0, v1, q, z1, r);

  dim3 gSq(UNITS / 64, UNITS / 64);       // (16,16), 64-thread blocks
  dim3 gWide(BATCH / 128, UNITS / 128);   // (64,8), 256-thread blocks

  // M = B^T B + 2I
  gemm_wmma<64, 64, false, true><<<gSq, 64, 0, stream>>>(
      Bt, B0, Mm, Mm, UNITS, UNITS, UNITS, 1.0f, 0.0f, 2.0f);
  // X0 = 0.137 * I
  init_inv_kernel<<<(unsigned)(UU / 256), 256, 0, stream>>>(X);

  // Newton-Schulz: X <- 2X - X (M X)
  for (int it = 0; it < 8; ++it) {
    gemm_wmma<64, 64, false, false><<<gSq, 64, 0, stream>>>(
        Mm, X, T, T, UNITS, UNITS, UNITS, 1.0f, 0.0f, 0.0f);
    gemm_wmma<64, 64, true, false><<<gSq, 64, 0, stream>>>(
        X, T, X, X2, UNITS, UNITS, UNITS, -1.0f, 2.0f, 0.0f);
    float* tmp = X; X = X2; X2 = tmp;
  }

  // r = z2 + B^T z1   (in place: each element read+written by the same thread)
  gemm_wmma<128, 128, true, false><<<gWide, 256, 0, stream>>>(
      Bt, z1, r, r, UNITS, BATCH, UNITS, 1.0f, 1.0f, 0.0f);
  // y = M^{-1} r
  gemm_wmma<128, 128, false, false><<<gWide, 256, 0, stream>>>(
      X, r, y, y, UNITS, BATCH, UNITS, 1.0f, 0.0f, 0.0f);
  // x = z1 - B y
  gemm_wmma<128, 128, true, false><<<gWide, 256, 0, stream>>>(
      B0, y, z1, x, UNITS, BATCH, UNITS, -1.0f, 1.0f, 0.0f);
}